// GCN_82231443849288
// MI455X (gfx1250) — compile-verified
//
#include <hip/hip_runtime.h>

#define NODES 100000
#define EDGES 1200000
#define F0 128
#define F1 256
#define F2 40

typedef float v2f __attribute__((ext_vector_type(2)));
typedef float v8f __attribute__((ext_vector_type(8)));

// ---------------------------------------------------------------- degrees ---
__global__ __launch_bounds__(256) void deg_init_kernel(float* __restrict__ deg) {
    int i = blockIdx.x * blockDim.x + threadIdx.x;
    if (i < NODES) deg[i] = 1.0f;  // self-loop contributes 1 to every node
}

__global__ __launch_bounds__(256) void deg_accum_kernel(const int* __restrict__ ei,
                                                        float* __restrict__ deg) {
    int e = blockIdx.x * blockDim.x + threadIdx.x;
    if (e < EDGES) {
        int c = ei[EDGES + e];  // col = edge_index[1]
        unsafeAtomicAdd(&deg[c], 1.0f);
    }
}

__global__ __launch_bounds__(256) void deg_rsqrt_kernel(float* __restrict__ deg) {
    int i = blockIdx.x * blockDim.x + threadIdx.x;
    if (i < NODES) deg[i] = rsqrtf(deg[i]);  // deg >= 1 always (self-loop)
}

// ------------------------------------------------------------------ GEMM1 ---
// h1[N,256] = x[N,128] @ W1[128,256]; one 16x16 tile per wave, K stepped by 4
// using V_WMMA_F32_16X16X4_F32.  A frag: lane(m=lane&15) holds K=kb,kb+1 with
// kb = k + (lane>>4)*2.  B frag mirrors with N instead of M.  C/D: VGPR v is
// row v + (lane>>4)*8, col = lane&15.
__global__ __launch_bounds__(128) void gemm1_wmma(const float* __restrict__ x,
                                                  const float* __restrict__ W,
                                                  float* __restrict__ h) {
    const int wave = threadIdx.x >> 5;
    const int lane = threadIdx.x & 31;
    const int half = lane >> 4;
    const int lm   = lane & 15;
    const int m0   = blockIdx.x * 16;
    const int n0   = (blockIdx.y * 4 + wave) * 16;

    const float* arow = x + (size_t)(m0 + lm) * F0 + half * 2;   // A: row-major
    const float* bcol = W + (size_t)(half * 2) * F1 + (n0 + lm); // B: row-major

    v8f c = {0.f, 0.f, 0.f, 0.f, 0.f, 0.f, 0.f, 0.f};
#pragma unroll
    for (int k = 0; k < F0; k += 4) {
        v2f a;
        a.x = arow[k];
        a.y = arow[k + 1];
        v2f b;
        b.x = bcol[(size_t)k * F1];
        b.y = bcol[(size_t)(k + 1) * F1];
        c = __builtin_amdgcn_wmma_f32_16x16x4_f32(false, a, false, b,
                                                  (short)0, c, false, false);
    }
    float* orow = h + (size_t)(m0 + half * 8) * F1 + n0 + lm;
#pragma unroll
    for (int v = 0; v < 8; ++v) orow[(size_t)v * F1] = c[v];
}

// ------------------------------------------------------------------ GEMM2 ---
// h2[N,40] = relu(agg1)[N,256] @ W2[256,40].  3 waves cover N tiles 0..2
// (48 cols); cols >= 40 are zero-padded on load and masked on store.
__global__ __launch_bounds__(96) void gemm2_wmma(const float* __restrict__ g,
                                                 const float* __restrict__ W,
                                                 float* __restrict__ h) {
    const int wave = threadIdx.x >> 5;
    const int lane = threadIdx.x & 31;
    const int half = lane >> 4;
    const int lm   = lane & 15;
    const int m0   = blockIdx.x * 16;
    const int n    = wave * 16 + lm;
    const bool nok = (n < F2);

    const float* arow = g + (size_t)(m0 + lm) * F1 + half * 2;

    v8f c = {0.f, 0.f, 0.f, 0.f, 0.f, 0.f, 0.f, 0.f};
#pragma unroll
    for (int k = 0; k < F1; k += 4) {
        v2f a;
        a.x = fmaxf(arow[k], 0.0f);      // fused relu
        a.y = fmaxf(arow[k + 1], 0.0f);
        v2f b;
        b.x = nok ? W[(size_t)(k + half * 2) * F2 + n] : 0.0f;
        b.y = nok ? W[(size_t)(k + half * 2 + 1) * F2 + n] : 0.0f;
        c = __builtin_amdgcn_wmma_f32_16x16x4_f32(false, a, false, b,
                                                  (short)0, c, false, false);
    }
    if (nok) {
        float* orow = h + (size_t)(m0 + half * 8) * F2 + n;
#pragma unroll
        for (int v = 0; v < 8; ++v) orow[(size_t)v * F2] = c[v];
    }
}

// ------------------------------------------------------------- aggregation ---
// out[i,f] = bias[f] + h[i,f] * dinv[i]^2     (self-loop term folded in)
template <int F>
__global__ __launch_bounds__(256) void agg_init_kernel(const float* __restrict__ h,
                                                       const float* __restrict__ bias,
                                                       const float* __restrict__ dinv,
                                                       float* __restrict__ out) {
    long long t = (long long)blockIdx.x * blockDim.x + threadIdx.x;
    if (t >= (long long)NODES * F) return;
    int i = (int)(t / F);
    int f = (int)(t % F);
    float d = dinv[i];
    out[t] = bias[f] + h[t] * d * d;
}

// out[col[e], f..f+3] += h[row[e], f..f+3] * dinv[row]*dinv[col]
template <int F>
__global__ __launch_bounds__(256) void agg_edges_kernel(const int* __restrict__ ei,
                                                        const float* __restrict__ h,
                                                        const float* __restrict__ dinv,
                                                        float* __restrict__ out) {
    constexpr int TPE = F / 4;  // threads per edge, 4 feats each
    long long t = (long long)blockIdx.x * blockDim.x + threadIdx.x;
    if (t >= (long long)EDGES * TPE) return;
    int e = (int)(t / TPE);
    int f = (int)(t % TPE) * 4;
    int r = ei[e];
    int c = ei[EDGES + e];
    float w = dinv[r] * dinv[c];
    const float4 v = *(const float4*)(h + (size_t)r * F + f);
    float* o = out + (size_t)c * F + f;
    unsafeAtomicAdd(o + 0, v.x * w);
    unsafeAtomicAdd(o + 1, v.y * w);
    unsafeAtomicAdd(o + 2, v.z * w);
    unsafeAtomicAdd(o + 3, v.w * w);
}

// ------------------------------------------------------------------ launch ---
extern "C" void kernel_launch(void* const* d_in, const int* in_sizes, int n_in,
                              void* d_out, int out_size, void* d_ws, size_t ws_size,
                              hipStream_t stream) {
    (void)in_sizes; (void)n_in; (void)out_size; (void)ws_size;
    const float* x  = (const float*)d_in[0];
    const int*   ei = (const int*)d_in[1];   // [2,E]: row = ei[0:E], col = ei[E:2E]
    const float* W1 = (const float*)d_in[2];
    const float* b1 = (const float*)d_in[3];
    const float* W2 = (const float*)d_in[4];
    const float* b2 = (const float*)d_in[5];
    float* out = (float*)d_out;

    char*  ws  = (char*)d_ws;
    size_t off = 0;
    auto carve = [&](size_t bytes) -> float* {
        float* p = (float*)(ws + off);
        off += (bytes + 255) & ~(size_t)255;
        return p;
    };
    float* dinv = carve((size_t)NODES * sizeof(float));
    float* h1   = carve((size_t)NODES * F1 * sizeof(float));
    float* agg1 = carve((size_t)NODES * F1 * sizeof(float));
    float* h2   = carve((size_t)NODES * F2 * sizeof(float));

    // ---- normalization ----
    deg_init_kernel<<<(NODES + 255) / 256, 256, 0, stream>>>(dinv);
    deg_accum_kernel<<<(EDGES + 255) / 256, 256, 0, stream>>>(ei, dinv);
    deg_rsqrt_kernel<<<(NODES + 255) / 256, 256, 0, stream>>>(dinv);

    // ---- layer 1 ----
    gemm1_wmma<<<dim3(NODES / 16, F1 / 64), 128, 0, stream>>>(x, W1, h1);
    {
        long long tot = (long long)NODES * F1;
        agg_init_kernel<F1><<<(unsigned)((tot + 255) / 256), 256, 0, stream>>>(h1, b1, dinv, agg1);
        long long te = (long long)EDGES * (F1 / 4);
        agg_edges_kernel<F1><<<(unsigned)((te + 255) / 256), 256, 0, stream>>>(ei, h1, dinv, agg1);
    }

    // ---- layer 2 ----
    gemm2_wmma<<<NODES / 16, 96, 0, stream>>>(agg1, W2, h2);
    {
        long long tot = (long long)NODES * F2;
        agg_init_kernel<F2><<<(unsigned)((tot + 255) / 256), 256, 0, stream>>>(h2, b2, dinv, out);
        long long te = (long long)EDGES * (F2 / 4);
        agg_edges_kernel<F2><<<(unsigned)((te + 255) / 256), 256, 0, stream>>>(ei, h2, dinv, out);
    }
}